// DownsampleWithHybridSampler_5471788335640
// MI455X (gfx1250) — compile-verified
//
#include <hip/hip_runtime.h>
#include <math.h>

// Problem constants (match reference)
#define BB     8
#define CC     64
#define NPTS   8192
#define MM     2048
#define KNN    16
#define FPS_M  1433          // int(2048 * 0.7)
#define TAU    0.2f

typedef __attribute__((ext_vector_type(2))) float v2f;
typedef __attribute__((ext_vector_type(8))) float v8f;

// ---------------------------------------------------------------------------
// Kernel 1: Farthest point sampling. One 1024-thread block per batch.
// Each thread owns 8 points (coords + running min-dist in registers).
// The argmax reduction carries (val, idx, x, y, z) so the winner's
// coordinates are produced by the reduction itself -> only 2 barriers
// per sequential iteration (the latency floor of this stage).
// ---------------------------------------------------------------------------
__global__ void fps_kernel(const float* __restrict__ pos, int* __restrict__ idx)
{
    const int b   = blockIdx.x;
    const int tid = threadIdx.x;              // 0..1023
    const float* px = pos + (size_t)b * 3 * NPTS;
    const float* py = px + NPTS;
    const float* pz = px + 2 * NPTS;

    float lx[8], ly[8], lz[8], ld[8];
#pragma unroll
    for (int i = 0; i < 8; ++i) {
        int n = tid + i * 1024;               // coalesced ownership
        lx[i] = px[n]; ly[i] = py[n]; lz[i] = pz[n];
        ld[i] = 1e10f;
    }

    __shared__ float wv[32], wx[32], wy[32], wz[32];
    __shared__ int   wi[32];
    __shared__ float scx, scy, scz;

    // deterministic start (reference uses jax PRNG; unreproducible bit-exactly)
    int start = (int)((2654435761u * (unsigned)(b + 1)) & (NPTS - 1));
    if (tid == 0) idx[b * MM + 0] = start;
    if (tid == (start & 1023)) {
        int ii = start >> 10;
        scx = lx[ii]; scy = ly[ii]; scz = lz[ii];
    }
    __syncthreads();

    for (int iter = 1; iter < FPS_M; ++iter) {
        float cx = scx, cy = scy, cz = scz;

        float bestv = -1.0f, bx = 0.f, by = 0.f, bz = 0.f;
        int   besti = 0;
#pragma unroll
        for (int i = 0; i < 8; ++i) {
            float dx = lx[i] - cx, dy = ly[i] - cy, dz = lz[i] - cz;
            float d  = fmaf(dx, dx, fmaf(dy, dy, dz * dz));
            ld[i] = fminf(ld[i], d);
            if (ld[i] > bestv) {
                bestv = ld[i]; besti = tid + i * 1024;
                bx = lx[i]; by = ly[i]; bz = lz[i];
            }
        }
        // wave32 reduce of the 5-tuple
#pragma unroll
        for (int off = 16; off > 0; off >>= 1) {
            float ov = __shfl_xor(bestv, off, 32);
            int   oi = __shfl_xor(besti, off, 32);
            float ox = __shfl_xor(bx, off, 32);
            float oy = __shfl_xor(by, off, 32);
            float oz = __shfl_xor(bz, off, 32);
            if (ov > bestv || (ov == bestv && oi < besti)) {
                bestv = ov; besti = oi; bx = ox; by = oy; bz = oz;
            }
        }
        if ((tid & 31) == 0) {
            int w = tid >> 5;
            wv[w] = bestv; wi[w] = besti; wx[w] = bx; wy[w] = by; wz[w] = bz;
        }
        __syncthreads();
        if (tid < 32) {
            bestv = wv[tid]; besti = wi[tid];
            bx = wx[tid]; by = wy[tid]; bz = wz[tid];
#pragma unroll
            for (int off = 16; off > 0; off >>= 1) {
                float ov = __shfl_xor(bestv, off, 32);
                int   oi = __shfl_xor(besti, off, 32);
                float ox = __shfl_xor(bx, off, 32);
                float oy = __shfl_xor(by, off, 32);
                float oz = __shfl_xor(bz, off, 32);
                if (ov > bestv || (ov == bestv && oi < besti)) {
                    bestv = ov; besti = oi; bx = ox; by = oy; bz = oz;
                }
            }
            if (tid == 0) {
                idx[b * MM + iter] = besti;
                scx = bx; scy = by; scz = bz;
            }
        }
        __syncthreads();
    }
}

// ---------------------------------------------------------------------------
// Kernel 2: fill random tail of the hybrid index set + gather pos_sub [B,3,m]
// ---------------------------------------------------------------------------
__global__ void sample_gather_kernel(const float* __restrict__ pos,
                                     const int* __restrict__ idx,
                                     float* __restrict__ psub)
{
    int t = blockIdx.x * blockDim.x + threadIdx.x;
    if (t >= BB * MM) return;
    int b = t / MM, j = t % MM;

    int n;
    if (j < FPS_M) {
        n = idx[b * MM + j];
    } else {
        unsigned h = (unsigned)j * 2654435761u ^ (unsigned)b * 0x9E3779B9u ^ 0xA511E9B3u;
        h ^= h >> 15; h *= 0x2C1B3C6Du; h ^= h >> 12; h *= 0x297A2D39u; h ^= h >> 15;
        n = (int)(h & (NPTS - 1));
    }
    const float* pb = pos + (size_t)b * 3 * NPTS;
    float* pc = psub + (size_t)b * 3 * MM;
    pc[j]          = pb[n];
    pc[MM + j]     = pb[NPTS + n];
    pc[2 * MM + j] = pb[2 * NPTS + n];
}

// ---------------------------------------------------------------------------
// Kernel 3: KNN (top-16 smallest squared distances) via V_WMMA_F32_16X16X4_F32.
// Block = 256 threads (8 waves) per (batch, 16-centroid tile).
//   A (16x4): rows = centroid (x, y, z, |ps|^2)
//   B (4x16): cols = point (-2x, -2y, -2z, 1)
//   C init   = |pa_n|^2 broadcast per column
//   => D = |ps|^2 + |pa|^2 - 2 ps.pa  (the reference d2), one WMMA per tile.
// Point slabs (3 x CHUNK floats) are double-buffered into LDS with
// global_load_async_to_lds_b128 issued one chunk ahead (ASYNCcnt fenced),
// overlapping global fetch with WMMA + selection of the current chunk.
// Selection: 16 threads keep a running top-16 in registers, scanning the
// LDS d2 rows with b128 loads and a 4-wide min pre-filter.
// ---------------------------------------------------------------------------
#define CHUNK   512
#define NCHUNK  (NPTS / CHUNK)
#define SD2P    (CHUNK + 4)    // row stride (floats): 16B-aligned, bank-spread

__device__ __forceinline__ void stage_chunk(const float* __restrict__ pb, int n0,
                                            float* dst, int tid)
{
    if (tid < CHUNK / 4) {                    // 128 lanes, b128 each, 3 rows
        unsigned dof = (unsigned)(size_t)dst; // low 32 bits = LDS offset
#pragma unroll
        for (int r = 0; r < 3; ++r) {
            const float* g = pb + (size_t)r * NPTS + n0 + tid * 4;
            unsigned l = dof + (unsigned)((r * CHUNK + tid * 4) * 4);
            asm volatile("global_load_async_to_lds_b128 %0, %1, off"
                         :: "v"(l), "v"(g) : "memory");
        }
    }
}

__global__ void knn_kernel(const float* __restrict__ pos,
                           const float* __restrict__ psub,
                           int* __restrict__ knn)
{
    const int blk  = blockIdx.x;              // B * (MM/16)
    const int b    = blk / (MM / 16);
    const int mt   = blk % (MM / 16);
    const int tid  = threadIdx.x;
    const int lane = tid & 31;
    const int wave = tid >> 5;
    const bool hi  = lane >= 16;
    const int  l15 = lane & 15;

    __shared__ __align__(16) float sd2[16][SD2P];
    __shared__ __align__(16) float sbuf[2][3][CHUNK];

    // ---- A operand (same 16 centroids for the whole block) ----
    const int j = mt * 16 + l15;
    const float* pc = psub + (size_t)b * 3 * MM;
    float ax = pc[j], ay = pc[MM + j], az = pc[2 * MM + j];
    float asq = ax * ax + ay * ay + az * az;
    v2f A;
    A.x = hi ? az  : ax;                      // K = 2 : 0
    A.y = hi ? asq : ay;                      // K = 3 : 1

    const float* pb = pos + (size_t)b * 3 * NPTS;

    // ---- running top-16 state (threads 0..15 only) ----
    float bd[16]; int bi[16];
#pragma unroll
    for (int k = 0; k < 16; ++k) { bd[k] = 3.0e38f; bi[k] = 0; }
    float curmax = 3.0e38f; int maxpos = 0;

    // prologue: stage chunk 0
    stage_chunk(pb, 0, &sbuf[0][0][0], tid);
    asm volatile("s_wait_asynccnt 0x0" ::: "memory");
    __syncthreads();

    for (int c = 0; c < NCHUNK; ++c) {
        const int n0 = c * CHUNK;
        const float* cur = &sbuf[c & 1][0][0];
        if (c + 1 < NCHUNK)
            stage_chunk(pb, n0 + CHUNK, &sbuf[(c + 1) & 1][0][0], tid);

        // ---- WMMA phase: CHUNK/16 tiles, round-robin over 8 waves ----
        for (int t = wave; t < CHUNK / 16; t += 8) {
            int nl = t * 16 + l15;            // index within chunk
            float x = cur[nl];
            float y = cur[CHUNK + nl];
            float z = cur[2 * CHUNK + nl];
            float sq = x * x + y * y + z * z;
            v2f Bv;
            Bv.x = hi ? (-2.0f * z) : (-2.0f * x);   // K = 2 : 0
            Bv.y = hi ? 1.0f        : (-2.0f * y);   // K = 3 : 1
            v8f Cv;
#pragma unroll
            for (int r = 0; r < 8; ++r) Cv[r] = sq;  // |pa_n|^2 per column

            Cv = __builtin_amdgcn_wmma_f32_16x16x4_f32(
                    false, A, false, Bv, (short)0, Cv, false, false);

            int rbase = hi ? 8 : 0;                  // C rows: VGPR r -> M=r / r+8
#pragma unroll
            for (int r = 0; r < 8; ++r) sd2[rbase + r][nl] = Cv[r];
        }
        __syncthreads();

        // ---- selection: thread t owns centroid row t, b128 scans ----
        if (tid < 16) {
            const float4* row = (const float4*)&sd2[tid][0];
            for (int i = 0; i < CHUNK / 4; ++i) {
                float4 v4 = row[i];
                float mn = fminf(fminf(v4.x, v4.y), fminf(v4.z, v4.w));
                if (mn < curmax) {
                    float vv[4] = { v4.x, v4.y, v4.z, v4.w };
#pragma unroll
                    for (int q = 0; q < 4; ++q) {
                        float v = vv[q];
                        if (v < curmax) {
                            bd[maxpos] = v; bi[maxpos] = n0 + i * 4 + q;
                            float nm = -1.0f; int np = 0;
#pragma unroll
                            for (int k = 0; k < 16; ++k)
                                if (bd[k] > nm) { nm = bd[k]; np = k; }
                            curmax = nm; maxpos = np;
                        }
                    }
                }
            }
        }
        // next buffer staged + sd2 consumed before the next WMMA phase
        asm volatile("s_wait_asynccnt 0x0" ::: "memory");
        __syncthreads();
    }

    if (tid < 16) {
        int base = ((b * MM) + (mt * 16 + tid)) * KNN;
#pragma unroll
        for (int k = 0; k < 16; ++k) knn[base + k] = bi[k];
    }
}

// ---------------------------------------------------------------------------
// Kernel 4: softmax(-d/tau) weights + weighted channel gather.
// Block = 64 threads per (b, j): threads 0..15 compute distances, thread 0
// does the 16-wide softmax in LDS, then each thread handles one channel c.
// x (16.7 MB) is L2-resident (192 MB L2) so the gathers stay on-chip.
// Sum over K -> independent of top-K ordering.
// ---------------------------------------------------------------------------
__global__ void agg_kernel(const float* __restrict__ x,
                           const float* __restrict__ pos,
                           const float* __restrict__ psub,
                           const int* __restrict__ knn,
                           float* __restrict__ xsub)
{
    const int bj = blockIdx.x;                // B * MM
    const int b  = bj / MM;
    const int j  = bj % MM;
    const int t  = threadIdx.x;

    __shared__ float w[16];
    __shared__ int   ni[16];

    if (t < 16) {
        int n = knn[(size_t)bj * KNN + t];
        ni[t] = n;
        const float* pb = pos  + (size_t)b * 3 * NPTS;
        const float* pc = psub + (size_t)b * 3 * MM;
        float dx = pb[n]            - pc[j];
        float dy = pb[NPTS + n]     - pc[MM + j];
        float dz = pb[2 * NPTS + n] - pc[2 * MM + j];
        float d  = sqrtf(fmaf(dx, dx, fmaf(dy, dy, dz * dz)));
        d = fmaxf(d, 1e-6f);
        w[t] = -d / TAU;
    }
    __syncthreads();
    if (t == 0) {
        float mx = w[0];
#pragma unroll
        for (int k = 1; k < 16; ++k) mx = fmaxf(mx, w[k]);
        float s = 0.0f;
#pragma unroll
        for (int k = 0; k < 16; ++k) { float e = __expf(w[k] - mx); w[k] = e; s += e; }
        float inv = 1.0f / s;
#pragma unroll
        for (int k = 0; k < 16; ++k) w[k] *= inv;
    }
    __syncthreads();

    const float* xb = x + ((size_t)b * CC + t) * NPTS;
    float acc = 0.0f;
#pragma unroll
    for (int k = 0; k < 16; ++k) acc = fmaf(xb[ni[k]], w[k], acc);
    xsub[((size_t)b * CC + t) * MM + j] = acc;
}

// ---------------------------------------------------------------------------
extern "C" void kernel_launch(void* const* d_in, const int* in_sizes, int n_in,
                              void* d_out, int out_size, void* d_ws, size_t ws_size,
                              hipStream_t stream)
{
    const float* x   = (const float*)d_in[0];   // [B, C, N]
    const float* pos = (const float*)d_in[1];   // [B, 3, N]
    // d_in[2] = m (compile-time constant MM)

    float* xsub = (float*)d_out;                       // [B, C, M]
    float* psub = xsub + (size_t)BB * CC * MM;         // [B, 3, M]

    int* idx = (int*)d_ws;                             // [B, M]
    int* knn = idx + BB * MM;                          // [B, M, K]

    fps_kernel<<<BB, 1024, 0, stream>>>(pos, idx);
    sample_gather_kernel<<<(BB * MM + 255) / 256, 256, 0, stream>>>(pos, idx, psub);
    knn_kernel<<<BB * (MM / 16), 256, 0, stream>>>(pos, psub, knn);
    agg_kernel<<<BB * MM, 64, 0, stream>>>(x, pos, psub, knn, xsub);
}